// JosephProjector3D_29214367547473
// MI455X (gfx1250) — compile-verified
//
#include <hip/hip_runtime.h>
#include <math.h>

typedef float v2f __attribute__((ext_vector_type(2)));
typedef float v8f __attribute__((ext_vector_type(8)));

#define X0   256      // detector samples / output size
#define ANG  360      // input angles
#define ZS   32       // slices
#define A2   180      // deduped angles
#define ND   363      // ceil(sqrt(2)*256): padded detector length
#define PADB 53       // pad_before = 363//2 - 256//2
#define NPC  192      // angle columns padded to 12*16 for WMMA
#define REC2 (ND*ND)  // 131769 recon pixels per slice
#define FROW 368      // padded detector row: [0]=guard(i=-1), [1..363]=i, [364..367]=guards
#define FPAD (A2*FROW) // 66240 floats per transposed filtered slice

// workspace layout (float offsets)
#define WS_TRIG 0                        // cos[180], sin[180]
#define WS_GK   512                      // g[0..319] analytic ramp kernel
#define WS_D    1024                     // deduped sino [32][256][192]
#define WS_F    (WS_D + X0*NPC*ZS)       // filtered sino, transposed [32][180][368]
#define WS_R    (WS_F + FPAD*ZS)         // recon [32][363][363]

#define AS1 __attribute__((address_space(1)))
#define AS3 __attribute__((address_space(3)))

#if defined(__HIP_DEVICE_COMPILE__) && \
    __has_builtin(__builtin_amdgcn_global_load_async_to_lds_b128) && \
    __has_builtin(__builtin_amdgcn_s_wait_asynccnt)
#define HAVE_ASYNC_LDS 1
#else
#define HAVE_ASYNC_LDS 0
#endif

#if HAVE_ASYNC_LDS
// GCC-style int4 vector: matches the builtin's V4i pointee exactly.
typedef int v4i __attribute__((vector_size(16)));

// generic->AS1: global generic addresses are the raw 64-bit address.
__device__ __forceinline__ AS1 v4i* glb_v4(const void* p) {
    return (AS1 v4i*)(unsigned long long)p;
}
// generic->AS3: LDS generic addresses carry the LDS offset in the low 32 bits.
__device__ __forceinline__ AS3 v4i* lds_v4(void* p) {
    return (AS3 v4i*)(unsigned)(unsigned long long)p;
}
#endif

// ---------------- K1: analytic tables ----------------
__global__ void fbp_tables(float* __restrict__ trig, float* __restrict__ gk) {
    int t = threadIdx.x;
    if (t < A2) {
        float th = (float)t * 0.017453292519943295f;  // pi/180
        trig[t]        = cosf(th);
        trig[t + A2]   = sinf(th);
    }
    if (t < 320) {
        float g;
        if (t == 0)            g = 0.5f;
        else if (t & 1) { float pk = 3.14159265358979f * (float)t; g = -2.0f / (pk * pk); }
        else                   g = 0.0f;
        gk[t] = g;
    }
}

// ---------------- K2: 360->180 dedup + zero-pad angle columns ----------------
// D[z][x][a] = 0.5*(s[x,a,z] + s[255-x,a+180,z]),  a<180; zeros for a in [180,192)
__global__ void fbp_dedup(const float* __restrict__ s, float* __restrict__ D) {
    int idx = blockIdx.x * blockDim.x + threadIdx.x;     // 32*256*192 threads
    int a = idx % NPC;
    int x = (idx / NPC) % X0;
    int z = idx / (NPC * X0);
    float v = 0.0f;
    if (a < A2) {
        float s0 = s[((size_t)x * ANG + a) * ZS + z];
        float s1 = s[((size_t)(X0 - 1 - x) * ANG + (a + A2)) * ZS + z];
        v = 0.5f * (s0 + s1);
    }
    D[(size_t)z * (X0 * NPC) + x * NPC + a] = v;
}

// ---------------- K2b: zero the guard cells of the transposed F rows ----------------
__global__ void fbp_fzero(float* __restrict__ F) {
    int idx = blockIdx.x * blockDim.x + threadIdx.x;     // ZS*A2 rows
    if (idx >= ZS * A2) return;
    int z = idx / A2, a = idx % A2;
    float* row = F + (size_t)z * FPAD + a * FROW;
    row[0] = 0.0f;                                       // i = -1 guard
    row[364] = 0.0f; row[365] = 0.0f; row[366] = 0.0f; row[367] = 0.0f;  // i>=363 guards
}

// ---------------- K3: ramp filter as WMMA f32 GEMM (transposed output) ----------------
// F[z][a][i] = sum_x D[z][x][a] * g(|i-53-x|)   i.e.  F^T = D^T (M=angle) x K^T (N=detector)
__global__ __launch_bounds__(32)
void fbp_filter_wmma(const float* __restrict__ gk, const float* __restrict__ D,
                     float* __restrict__ F) {
    int tm = blockIdx.x;            // 12 angle tiles (192, 180 valid)
    int tn = blockIdx.y;            // 23 detector tiles (368, 363 valid)
    int z  = blockIdx.z;
    int L  = threadIdx.x;           // lane 0..31 (wave32)
    const float* Dz = D + (size_t)z * (X0 * NPC);

    int rowA  = tm * 16 + (L & 15);          // A row (M = angle); D padded to 192 cols
    int n     = tn * 16 + (L & 15);          // B col (N = detector i)
    int khalf = (L >> 4) * 2;                // lanes 16-31 hold K+2,K+3

    v8f c = {};
    #pragma unroll 4
    for (int k4 = 0; k4 < X0 / 4; ++k4) {
        int kb = k4 * 4 + khalf;
        v2f a, b;
        a.x = Dz[kb * NPC + rowA];           // A[m][k] = D[k][m]
        a.y = Dz[(kb + 1) * NPC + rowA];
        int d0 = n - PADB - kb;              // B[k][n] = g(|n-53-k|), |d| <= 314 < 320
        int d1 = d0 - 1;
        b.x = gk[d0 < 0 ? -d0 : d0];
        b.y = gk[d1 < 0 ? -d1 : d1];
        // (neg_a, A, neg_b, B, c_mod, C, reuse_a, reuse_b)
        c = __builtin_amdgcn_wmma_f32_16x16x4_f32(false, a, false, b,
                                                  (short)0, c, false, false);
    }

    // C elem r -> angle tm*16 + r + (lane>=16 ? 8 : 0), detector tn*16 + (lane&15)
    int col = tn * 16 + (L & 15);
    int rbase = tm * 16 + ((L >> 4) ? 8 : 0);
    float* Fz = F + (size_t)z * FPAD;
    #pragma unroll
    for (int r = 0; r < 8; ++r) {
        int row = rbase + r;
        if (row < A2 && col < ND) Fz[row * FROW + 1 + col] = c[r];   // coalesced across lanes
    }
}

// ---------------- K4: backprojection from LDS-resident transposed slice ----------------
// 320KB WGP LDS holds the full [180][368] slice (265KB) + trig. Async DMA fill.
// In-circle pixels need no bounds checks (|t|<=181); guards absorb rim rounding.
__global__ __launch_bounds__(256)
void fbp_backproject(const float* __restrict__ F, const float* __restrict__ trig,
                     float* __restrict__ R) {
    extern __shared__ float lds[];
    float* Fl = lds;                 // [180][368] incl. guards
    float* ct = lds + FPAD;          // [180]
    float* st = ct + A2;             // [180]
    int z   = blockIdx.y;
    int tid = threadIdx.x;

    const float* Fz = F + (size_t)z * FPAD;
#if HAVE_ASYNC_LDS
    // FPAD = 66240 is a multiple of 4; buffers are 16B aligned -> b128 DMA.
    for (int i = 4 * tid; i < FPAD; i += 4 * 256)
        __builtin_amdgcn_global_load_async_to_lds_b128(glb_v4(Fz + i), lds_v4(Fl + i), 0, 0);
    __builtin_amdgcn_s_wait_asynccnt(0);
#else
    for (int i = tid; i < FPAD; i += 256) Fl[i] = Fz[i];
#endif
    for (int i = tid; i < A2; i += 256) { ct[i] = trig[i]; st[i] = trig[i + A2]; }
    __syncthreads();

    const float scale = 0.0087266462599716477f;   // pi / 360
    float* Rz = R + (size_t)z * REC2;
    int stride = gridDim.x * 256;
    for (int p = blockIdx.x * 256 + tid; p < REC2; p += stride) {
        int py = p / ND, px = p % ND;
        int iyy = py - 181, ixx = px - 181;
        if (ixx * ixx + iyy * iyy > 181 * 181) { Rz[p] = 0.0f; continue; }
        float yy = (float)iyy, xx = (float)ixx;
        float acc = 0.0f;
        #pragma unroll 4
        for (int a = 0; a < A2; ++a) {
            float iy = fmaf(yy, ct[a], fmaf(-xx, st[a], 181.0f));   // in [0,362] +- eps
            float fl = floorf(iy);
            float fr = iy - fl;
            int   i0 = (int)fl;                                     // in [-1, 362]
            const float* q = Fl + (a * FROW + 1) + i0;              // adjacent taps
            float v0 = q[0], v1 = q[1];
            acc += v0 + fr * (v1 - v0);
        }
        Rz[p] = acc * scale;
    }
}

// ---------------- K5: bilinear resize 363->256 + transpose into output ----------------
__device__ __forceinline__ void rs_idx(int o, int& i0, int& i1, float& w) {
    float src = fmaxf(((float)o + 0.5f) * (363.0f / 256.0f) - 0.5f, 0.0f);
    i0 = (int)src; if (i0 > ND - 1) i0 = ND - 1;
    i1 = i0 + 1;   if (i1 > ND - 1) i1 = ND - 1;
    w = src - (float)i0;
}

__global__ void fbp_resize(const float* __restrict__ R, float* __restrict__ out) {
    int idx = blockIdx.x * blockDim.x + threadIdx.x;   // 256*256*32
    int z = idx & 31;
    int y = (idx >> 5) & 255;
    int x = idx >> 13;
    int r0, r1, c0, c1; float wr, wc;
    rs_idx(y, r0, r1, wr);    // output (x,y) samples resized[y][x]  (transpose)
    rs_idx(x, c0, c1, wc);
    const float* Rz = R + (size_t)z * REC2;
    float top = Rz[r0 * ND + c0] * (1.0f - wc) + Rz[r0 * ND + c1] * wc;
    float bot = Rz[r1 * ND + c0] * (1.0f - wc) + Rz[r1 * ND + c1] * wc;
    out[idx] = top * (1.0f - wr) + bot * wr;
}

extern "C" void kernel_launch(void* const* d_in, const int* in_sizes, int n_in,
                              void* d_out, int out_size, void* d_ws, size_t ws_size,
                              hipStream_t stream) {
    (void)in_sizes; (void)n_in; (void)out_size; (void)ws_size;
    const float* sino = (const float*)d_in[0];
    float* out  = (float*)d_out;
    float* ws   = (float*)d_ws;
    float* trig = ws + WS_TRIG;
    float* gk   = ws + WS_GK;
    float* D    = ws + WS_D;
    float* F    = ws + WS_F;
    float* R    = ws + WS_R;

    fbp_tables<<<1, 512, 0, stream>>>(trig, gk);

    fbp_dedup<<<(ZS * X0 * NPC) / 256, 256, 0, stream>>>(sino, D);

    fbp_fzero<<<(ZS * A2 + 255) / 256, 256, 0, stream>>>(F);

    dim3 gGemm(12, 23, ZS);      // (angle tiles, detector tiles, slices)
    fbp_filter_wmma<<<gGemm, 32, 0, stream>>>(gk, D, F);

    size_t shmem = (size_t)(FPAD + 2 * A2) * sizeof(float);   // 266,400 B < 320 KB
    (void)hipFuncSetAttribute((const void*)fbp_backproject,
                              hipFuncAttributeMaxDynamicSharedMemorySize, (int)shmem);
    dim3 gBp(8, ZS);
    fbp_backproject<<<gBp, 256, shmem, stream>>>(F, trig, R);

    fbp_resize<<<(X0 * X0 * ZS) / 256, 256, 0, stream>>>(R, out);
}